// QBatchNorm2d_76330158785277
// MI455X (gfx1250) — compile-verified
//
#include <hip/hip_runtime.h>

typedef __attribute__((ext_vector_type(2))) float v2f;
typedef __attribute__((ext_vector_type(4))) float f4;
typedef __attribute__((ext_vector_type(8))) float v8f;

#define Bn 32
#define Cn 256
#define Hn 56
#define Wn 56
#define CQ 64
#define HWn (Hn * Wn)          /* 3136  */
#define CSTRIDE (64 * HWn)     /* component stride in floats: 200704 */
#define IMGSTRIDE (Cn * HWn)   /* per-batch stride: 802816 */
#define NTOT (Bn * HWn)        /* samples per statistic: 100352 */
#define EPSV 1e-5f

// ---------------------------------------------------------------------------
// Pass 1: per-(group, batch) partial sums: 4 component sums + 10 cross sums.
// One block = one (c, b) pair; float4 (b128) loads of the 4 component planes.
// ---------------------------------------------------------------------------
__global__ __launch_bounds__(256) void qbn_stats(const float* __restrict__ x,
                                                 float* __restrict__ partial) {
    const int c = blockIdx.x, b = blockIdx.y;
    const int tid = threadIdx.x;
    const float* p0 = x + (size_t)b * IMGSTRIDE + (size_t)c * HWn;
    const f4* q0 = (const f4*)(p0);
    const f4* q1 = (const f4*)(p0 + CSTRIDE);
    const f4* q2 = (const f4*)(p0 + 2 * (size_t)CSTRIDE);
    const f4* q3 = (const f4*)(p0 + 3 * (size_t)CSTRIDE);

    float acc[14];
#pragma unroll
    for (int k = 0; k < 14; ++k) acc[k] = 0.f;

    const int nvec = HWn / 4;  // 784 float4s per plane
    for (int i = tid; i < nvec; i += 256) {
        f4 a = q0[i], bb = q1[i], cc = q2[i], dd = q3[i];
#pragma unroll
        for (int j = 0; j < 4; ++j) {
            float v0 = a[j], v1 = bb[j], v2 = cc[j], v3 = dd[j];
            acc[0] += v0; acc[1] += v1; acc[2] += v2; acc[3] += v3;
            acc[4]  += v0 * v0; acc[5]  += v0 * v1; acc[6]  += v0 * v2; acc[7]  += v0 * v3;
            acc[8]  += v1 * v1; acc[9]  += v1 * v2; acc[10] += v1 * v3;
            acc[11] += v2 * v2; acc[12] += v2 * v3; acc[13] += v3 * v3;
        }
    }

    // wave32 shuffle reduce, then LDS across the 8 waves of the block
    __shared__ float smem[8][14];
    const int lane = tid & 31, wid = tid >> 5;
#pragma unroll
    for (int k = 0; k < 14; ++k) {
        float v = acc[k];
        for (int o = 16; o > 0; o >>= 1) v += __shfl_down(v, o, 32);
        if (lane == 0) smem[wid][k] = v;
    }
    __syncthreads();
    if (wid == 0) {
#pragma unroll
        for (int k = 0; k < 14; ++k) {
            float v = (lane < 8) ? smem[lane][k] : 0.f;
            for (int o = 4; o > 0; o >>= 1) v += __shfl_down(v, o, 32);
            if (lane == 0) partial[((size_t)c * Bn + b) * 14 + k] = v;
        }
    }
}

// ---------------------------------------------------------------------------
// Pass 2: per group (64 threads total): finalize mean/cov, 4x4 Cholesky,
// triangular inverse, compose  M = W * U^-1  and  v = bias - M * mean.
// Output layout per group c: mv[c*20 + p*4 + r] = M[p][r], mv[c*20+16+p] = v[p]
// ---------------------------------------------------------------------------
__global__ __launch_bounds__(64) void qbn_factor(const float* __restrict__ partial,
                                                 const float* __restrict__ weight,
                                                 const float* __restrict__ bias,
                                                 float* __restrict__ mv) {
    const int c = threadIdx.x;
    if (c >= CQ) return;

    float S[14];
#pragma unroll
    for (int k = 0; k < 14; ++k) S[k] = 0.f;
    for (int b = 0; b < Bn; ++b) {
        const float* p = partial + ((size_t)c * Bn + b) * 14;
#pragma unroll
        for (int k = 0; k < 14; ++k) S[k] += p[k];
    }

    const float invN = 1.f / (float)NTOT;
    float mean[4];
#pragma unroll
    for (int q = 0; q < 4; ++q) mean[q] = S[q] * invN;

    // covariance + eps*I (pairs order: 00,01,02,03,11,12,13,22,23,33)
    float A[4][4];
    int idx = 4;
    for (int q = 0; q < 4; ++q)
        for (int p = q; p < 4; ++p) {
            float cv = S[idx++] * invN - mean[q] * mean[p];
            A[q][p] = cv;
            A[p][q] = cv;
        }
    for (int q = 0; q < 4; ++q) A[q][q] += EPSV;

    // Cholesky: A = L L^T (lower)
    float L[4][4] = {{0.f}};
    for (int i = 0; i < 4; ++i) {
        float s = A[i][i];
        for (int k = 0; k < i; ++k) s -= L[i][k] * L[i][k];
        L[i][i] = sqrtf(fmaxf(s, 1e-20f));
        for (int j = i + 1; j < 4; ++j) {
            float t = A[j][i];
            for (int k = 0; k < i; ++k) t -= L[j][k] * L[i][k];
            L[j][i] = t / L[i][i];
        }
    }

    // inverse of lower-triangular L
    float Li[4][4] = {{0.f}};
    for (int i = 0; i < 4; ++i) {
        Li[i][i] = 1.f / L[i][i];
        for (int j = i + 1; j < 4; ++j) {
            float s = 0.f;
            for (int k = i; k < j; ++k) s += L[j][k] * Li[k][i];
            Li[j][i] = -s / L[j][j];
        }
    }

    // U^-1 = (L^T)^-1 = Li^T ;  M[p][r] = sum_q W[p][q] * Uinv[q][r]
    //                                   = sum_q W[p][q] * Li[r][q]
    float M[4][4];
    for (int p = 0; p < 4; ++p)
        for (int r = 0; r < 4; ++r) {
            float s = 0.f;
            for (int q = 0; q < 4; ++q)
                s += weight[((size_t)p * 4 + q) * CQ + c] * Li[r][q];
            M[p][r] = s;
        }

    float* g = mv + (size_t)c * 20;
    for (int p = 0; p < 4; ++p) {
        float vb = bias[(size_t)p * CQ + c];
        for (int r = 0; r < 4; ++r) {
            vb -= M[p][r] * mean[r];
            g[p * 4 + r] = M[p][r];
        }
        g[16 + p] = vb;
    }
}

// ---------------------------------------------------------------------------
// Pass 3: out = M*x + v via V_WMMA_F32_16X16X4_F32.
//   A(16x4)  = M rows replicated 4x :  A[m][k] = M[m%4][k]
//   B(4x16)  = 16 samples x 4 comps :  B[k][n] = x_k[s0+n]
//   C(16x16) = v broadcast          :  C[m][n] = v[m%4]
//   D row j (j<4), lanes 0..15      = out_j[s0 + lane]  -> coalesced stores
// ---------------------------------------------------------------------------
__global__ __launch_bounds__(256) void qbn_apply(const float* __restrict__ x,
                                                 const float* __restrict__ mv,
                                                 float* __restrict__ out) {
    const int c = blockIdx.x, b = blockIdx.y;
    const int tid = threadIdx.x;
    const int lane = tid & 31, wid = tid >> 5;
    const int n = lane & 15, half = lane >> 4;

    const float* g = mv + (size_t)c * 20;

    // A operand per ISA layout: lane holds A[n][2*half], A[n][2*half+1]
    v2f Amat;
    Amat.x = g[(n & 3) * 4 + 2 * half + 0];
    Amat.y = g[(n & 3) * 4 + 2 * half + 1];

    // C operand: VGPR j holds row (j or j+8) -> v[(row)%4] = v[j%4]
    v8f cin;
#pragma unroll
    for (int j = 0; j < 8; ++j) cin[j] = g[16 + (j & 3)];

    const size_t base = (size_t)b * IMGSTRIDE + (size_t)c * HWn;
    const float* x0 = x + base + (size_t)(2 * half) * CSTRIDE;
    const float* x1 = x + base + (size_t)(2 * half + 1) * CSTRIDE;
    float* o0 = out + base;

    const int ntiles = HWn / 16;  // 196 tiles of 16 samples per (c,b)
    for (int t = wid; t < ntiles; t += 8) {
        const int s0 = t * 16;

        // B operand: lane holds B[2*half][n], B[2*half+1][n]
        v2f Bmat;
        Bmat.x = x0[s0 + n];
        Bmat.y = x1[s0 + n];

        v8f d = __builtin_amdgcn_wmma_f32_16x16x4_f32(
            /*neg_a=*/false, Amat, /*neg_b=*/false, Bmat,
            /*c_mod=*/(short)0, cin, /*reuse_a=*/false, /*reuse_b=*/false);

        if (lane < 16) {
            o0[0 * (size_t)CSTRIDE + s0 + lane] = d[0];
            o0[1 * (size_t)CSTRIDE + s0 + lane] = d[1];
            o0[2 * (size_t)CSTRIDE + s0 + lane] = d[2];
            o0[3 * (size_t)CSTRIDE + s0 + lane] = d[3];
        }
    }
}

// ---------------------------------------------------------------------------
extern "C" void kernel_launch(void* const* d_in, const int* in_sizes, int n_in,
                              void* d_out, int out_size, void* d_ws, size_t ws_size,
                              hipStream_t stream) {
    const float* x      = (const float*)d_in[0];
    const float* weight = (const float*)d_in[1];
    const float* bias   = (const float*)d_in[2];
    float* out = (float*)d_out;

    float* ws      = (float*)d_ws;
    float* partial = ws;                      // CQ*Bn*14 = 28672 floats
    float* mv      = ws + (size_t)CQ * Bn * 14;  // CQ*20  = 1280 floats

    qbn_stats <<<dim3(CQ, Bn), 256, 0, stream>>>(x, partial);
    qbn_factor<<<1, 64, 0, stream>>>(partial, weight, bias, mv);
    qbn_apply <<<dim3(CQ, Bn), 256, 0, stream>>>(x, mv, out);
}